// SA_89721866813655
// MI455X (gfx1250) — compile-verified
//
#include <hip/hip_runtime.h>
#include <math.h>

// Self-attention (SAGAN): B=4, C=256, D=32, N=4096.
// Fused flash-attention, V_WMMA_F32_16X16X4_F32, double-buffered
// global_load_async_to_lds_b128 staging, DPP row_ror softmax reductions.
// Workspace requirement: 20 MB (q[B,N,32] + k/v pre-swizzled tiles, fp32).

typedef float v2f __attribute__((ext_vector_type(2)));
typedef float v4f __attribute__((ext_vector_type(4)));
typedef float v8f __attribute__((ext_vector_type(8)));

#define NTOK 4096
#define CCH  256
#define DQK  32

// DPP16 row-rotate move (rotates within 16-lane rows; pure VALU).
template <int CTRL>
__device__ __forceinline__ float dpp_rot(float x) {
  return __builtin_bit_cast(
      float, __builtin_amdgcn_update_dpp(0, __builtin_bit_cast(int, x), CTRL,
                                         0xf, 0xf, true));
}
// Reduce over the 16-lane row (rotation butterfly: offsets 1,2,4,8).
__device__ __forceinline__ float row_max16(float r) {
  r = fmaxf(r, dpp_rot<0x128>(r));   // row_ror:8
  r = fmaxf(r, dpp_rot<0x124>(r));   // row_ror:4
  r = fmaxf(r, dpp_rot<0x122>(r));   // row_ror:2
  r = fmaxf(r, dpp_rot<0x121>(r));   // row_ror:1
  return r;
}
__device__ __forceinline__ float row_sum16(float r) {
  r += dpp_rot<0x128>(r);
  r += dpp_rot<0x124>(r);
  r += dpp_rot<0x122>(r);
  r += dpp_rot<0x121>(r);
  return r;
}

// ---------------------------------------------------------------------------
// q/k projection. q stored [B,N,32] row-major (A-fragment friendly).
// k stored PRE-SWIZZLED per 16-key tile: intra-tile float index
//   p*32 + key*2 + comp   with d = 2p+comp
// so a 16B async copy lands directly in WMMA-B-fragment LDS layout.
// ---------------------------------------------------------------------------
__global__ __launch_bounds__(128) void proj_qk_kernel(
    const float* __restrict__ x, const float* __restrict__ y,
    const float* __restrict__ Wq, const float* __restrict__ bq,
    const float* __restrict__ Wk, const float* __restrict__ bk,
    float* __restrict__ qws, float* __restrict__ kws) {
  __shared__ float wqT[CCH * DQK];  // [c][d]
  __shared__ float wkT[CCH * DQK];
  const int tid = threadIdx.x;
  for (int i = tid; i < CCH * DQK; i += 128) {
    const int d = i >> 8, c = i & 255;  // i = d*256 + c (coalesced read)
    wqT[c * DQK + d] = Wq[i];
    wkT[c * DQK + d] = Wk[i];
  }
  __syncthreads();

  const int t = blockIdx.x * 128 + tid;  // 0 .. B*N-1
  const int b = t >> 12;
  const int n = t & (NTOK - 1);
  const float* xb = x + ((size_t)b * CCH) * NTOK + n;
  const float* yb = y + ((size_t)b * CCH) * NTOK + n;

  float qa[DQK], ka[DQK];
#pragma unroll
  for (int d = 0; d < DQK; ++d) { qa[d] = 0.f; ka[d] = 0.f; }

  for (int c = 0; c < CCH; ++c) {
    const float xv = xb[(size_t)c * NTOK];
    const float yv = yb[(size_t)c * NTOK];
    const float* wq = wqT + c * DQK;
    const float* wk = wkT + c * DQK;
#pragma unroll
    for (int d = 0; d < DQK; ++d) {
      qa[d] = fmaf(wq[d], xv, qa[d]);
      ka[d] = fmaf(wk[d], yv, ka[d]);
    }
  }
  float* qo = qws + (size_t)t * DQK;
  float* ko = kws + (size_t)(t & ~15) * DQK + (t & 15) * 2;  // swizzled tile
#pragma unroll
  for (int d = 0; d < DQK; ++d) {
    qo[d] = qa[d] + bq[d];
    ko[(d >> 1) * 32 + (d & 1)] = ka[d] + bk[d];
  }
}

// ---------------------------------------------------------------------------
// v projection, PRE-SWIZZLED per 16-key tile: intra-tile float index
//   p*512 + c*2 + comp   with key = 2p+comp  (c = output channel)
// ---------------------------------------------------------------------------
__global__ __launch_bounds__(128) void proj_v_kernel(
    const float* __restrict__ z, const float* __restrict__ Wv,
    const float* __restrict__ bv, float* __restrict__ vws) {
  __shared__ float wvT[CCH * 64];  // [c][e] for this e-tile, 64KB
  const int tid = threadIdx.x;
  const int et = blockIdx.x >> 7;  // 0..3
  const int nb = blockIdx.x & 127;
  const int e0 = et * 64;
  for (int i = tid; i < CCH * 64; i += 128) {
    const int e = i >> 8, c = i & 255;
    wvT[c * 64 + e] = Wv[(size_t)(e0 + e) * CCH + c];
  }
  __syncthreads();

  const int t = nb * 128 + tid;
  const int b = t >> 12;
  const int n = t & (NTOK - 1);
  const float* zb = z + ((size_t)b * CCH) * NTOK + n;

  float acc[64];
#pragma unroll
  for (int e = 0; e < 64; ++e) acc[e] = 0.f;

  for (int c = 0; c < CCH; ++c) {
    const float zv = zb[(size_t)c * NTOK];
    const float* wv = wvT + c * 64;
#pragma unroll
    for (int e = 0; e < 64; ++e) acc[e] = fmaf(wv[e], zv, acc[e]);
  }
  // swizzled store: tile base + p*512 + comp, element stride 2
  float* vo = vws + (size_t)(t & ~15) * CCH + ((t >> 1) & 7) * 512 + (t & 1);
#pragma unroll
  for (int e = 0; e < 64; ++e) vo[(size_t)(e0 + e) * 2] = acc[e] + bv[e0 + e];
}

// ---------------------------------------------------------------------------
// Fused flash-attention. 1 wave = 16 query rows, 2 waves/block, 512 blocks.
// Double-buffered async K/V tile staging; 8x WMMA QK^T + 64x WMMA PV per tile.
// ---------------------------------------------------------------------------
__global__ __launch_bounds__(64) void attn_kernel(
    const float* __restrict__ qws, const float* __restrict__ kws,
    const float* __restrict__ vws, const float* __restrict__ z,
    float* __restrict__ out) {
  const int tid = threadIdx.x;
  const int lane = tid & 31;
  const int w = tid >> 5;           // wave in block: 0..1
  const int lane16 = lane & 15;
  const int hi = (lane >> 4) & 1;   // half-wave id

  const int bIdx = blockIdx.x >> 7;  // 128 blocks per batch
  const int grp = blockIdx.x & 127;
  const int qBase = (grp * 2 + w) * 16;
  const size_t bN = (size_t)bIdx * NTOK;

  // 40KB LDS, manually partitioned; epilogue reuses it for O staging.
  __shared__ float smem[10240];
  float* kb0 = smem;                // 512 floats  (K tile, swizzled)
  float* kb1 = smem + 512;
  float* vb0 = smem + 1024;         // 4096 floats (V tile, swizzled)
  float* vb1 = smem + 5120;
  float* pw = smem + 9216 + w * 288;  // per-wave P tile, padded stride 18

  const float* kg = kws + bN * DQK;   // per-batch swizzled K base
  const float* vg = vws + bN * CCH;   // per-batch swizzled V base

  // Issue async copies of one 16-key tile (K: 2KB, V: 16KB) into LDS buffer.
  auto issue_tile = [&](int kt, float* kdst, float* vdst) {
#pragma unroll
    for (int it = 0; it < 2; ++it) {
      const int ch = tid + it * 64;  // 0..127 chunks of 16B
      const unsigned lds = (unsigned)(uintptr_t)(kdst + ch * 4);
      const unsigned goff = (unsigned)(((kt << 4) * DQK + ch * 4) * 4);
      asm volatile("global_load_async_to_lds_b128 %0, %1, %2"
                   :: "v"(lds), "v"(goff), "s"(kg) : "memory");
    }
#pragma unroll
    for (int it = 0; it < 16; ++it) {
      const int ch = tid + it * 64;  // 0..1023 chunks of 16B
      const unsigned lds = (unsigned)(uintptr_t)(vdst + ch * 4);
      const unsigned goff = (unsigned)(((kt << 4) * CCH + ch * 4) * 4);
      asm volatile("global_load_async_to_lds_b128 %0, %1, %2"
                   :: "v"(lds), "v"(goff), "s"(vg) : "memory");
    }
  };

  // Preload Q A-fragments (loop-invariant).
  v2f aQ[8];
  {
    const float* qrow = qws + (bN + qBase + lane16) * DQK + 2 * hi;
#pragma unroll
    for (int i = 0; i < 8; ++i) aQ[i] = *(const v2f*)(qrow + 4 * i);
  }

  const v8f zero8 = {0.f, 0.f, 0.f, 0.f, 0.f, 0.f, 0.f, 0.f};
  v8f o[16];
#pragma unroll
  for (int i = 0; i < 16; ++i) o[i] = zero8;

  float mrow[8], lrow[8];
#pragma unroll
  for (int j = 0; j < 8; ++j) { mrow[j] = -__builtin_inff(); lrow[j] = 0.f; }

  issue_tile(0, kb0, vb0);  // prologue

  for (int kt = 0; kt < NTOK / 16; ++kt) {
    asm volatile("s_wait_asynccnt 0x0" ::: "memory");
    __syncthreads();

    if (kt + 1 < NTOK / 16) {  // prefetch next tile into the other buffer
      if ((kt + 1) & 1) issue_tile(kt + 1, kb1, vb1);
      else              issue_tile(kt + 1, kb0, vb0);
    }
    const float* kcur = (kt & 1) ? kb1 : kb0;
    const float* vcur = (kt & 1) ? vb1 : vb0;

    // S = Q K^T (16x16), K-dim = 32 in 8 steps of 4
    v8f s = zero8;
#pragma unroll
    for (int k0i = 0; k0i < 8; ++k0i) {
      const int p = k0i * 2 + hi;
      const v2f bk = *(const v2f*)(kcur + p * 32 + lane16 * 2);
      s = __builtin_amdgcn_wmma_f32_16x16x4_f32(false, aQ[k0i], false, bk,
                                                (short)0, s, false, false);
    }

    // Online softmax: VGPR j of s holds row (j + hi*8), col = lane16.
    float alpha[8];
#pragma unroll
    for (int j = 0; j < 8; ++j) {
      const float r = row_max16(s[j]);
      const float mn = fmaxf(mrow[j], r);
      alpha[j] = __expf(mrow[j] - mn);
      mrow[j] = mn;
      const float pv = __expf(s[j] - mn);
      s[j] = pv;
      lrow[j] = lrow[j] * alpha[j] + row_sum16(pv);
    }

    // P -> LDS (wave-private, padded stride 18)
#pragma unroll
    for (int j = 0; j < 8; ++j) pw[(j + hi * 8) * 18 + lane16] = s[j];

    // Rescale accumulator
#pragma unroll
    for (int ct = 0; ct < 16; ++ct)
#pragma unroll
      for (int j = 0; j < 8; ++j) o[ct][j] *= alpha[j];

    // O += P (16x16) @ V (16x256): 4 K-steps x 16 column tiles
#pragma unroll
    for (int kk = 0; kk < 4; ++kk) {
      const int k0 = kk * 4;
      const v2f aP = *(const v2f*)(pw + lane16 * 18 + k0 + 2 * hi);
      const int p = (k0 >> 1) + hi;
      const float* vrow = vcur + p * 512;
#pragma unroll
      for (int ct = 0; ct < 16; ++ct) {
        const v2f bv2 = *(const v2f*)(vrow + (ct * 16 + lane16) * 2);
        o[ct] = __builtin_amdgcn_wmma_f32_16x16x4_f32(false, aP, false, bv2,
                                                      (short)0, o[ct], false,
                                                      false);
      }
    }
  }

  // Epilogue: normalize, stage O through LDS (reuse smem; stride 257).
  __syncthreads();  // all waves done reading K/V buffers
  float rinv[8];
#pragma unroll
  for (int j = 0; j < 8; ++j) rinv[j] = 1.0f / lrow[j];

  float* ow = smem + w * 4112;
#pragma unroll
  for (int ct = 0; ct < 16; ++ct)
#pragma unroll
    for (int j = 0; j < 8; ++j)
      ow[(j + hi * 8) * 257 + ct * 16 + lane16] = o[ct][j] * rinv[j];

  // m-coalesced writeback: out[b,c,m] = z[b,c,m] + O[m - qBase][c]
  for (int cc = 0; cc < CCH; cc += 2) {
    const int c = cc + hi;
    const float val = ow[lane16 * 257 + c];
    const size_t gidx = ((size_t)bIdx * CCH + c) * NTOK + qBase + lane16;
    out[gidx] = z[gidx] + val;
  }
}

// ---------------------------------------------------------------------------
extern "C" void kernel_launch(void* const* d_in, const int* in_sizes, int n_in,
                              void* d_out, int out_size, void* d_ws,
                              size_t ws_size, hipStream_t stream) {
  const float* z  = (const float*)d_in[0];
  const float* x  = (const float*)d_in[1];
  const float* y  = (const float*)d_in[2];
  const float* Wq = (const float*)d_in[3];
  const float* bq = (const float*)d_in[4];
  const float* Wk = (const float*)d_in[5];
  const float* bk = (const float*)d_in[6];
  const float* Wv = (const float*)d_in[7];
  const float* bv = (const float*)d_in[8];
  float* out = (float*)d_out;

  // Workspace (fp32): q[4,4096,32] | k swizzled | v swizzled  (20 MB total)
  float* ws = (float*)d_ws;
  float* qws = ws;
  float* kws = ws + 524288;
  float* vws = ws + 1048576;

  proj_qk_kernel<<<128, 128, 0, stream>>>(x, y, Wq, bq, Wk, bk, qws, kws);
  proj_v_kernel<<<512, 128, 0, stream>>>(z, Wv, bv, vws);
  attn_kernel<<<512, 64, 0, stream>>>(qws, kws, vws, z, out);
}